// SPHPhysicsInformedLoss_21715354649030
// MI455X (gfx1250) — compile-verified
//
#include <hip/hip_runtime.h>
#include <hip/hip_bf16.h>
#include <math.h>

typedef __attribute__((ext_vector_type(2))) float v2f;
typedef __attribute__((ext_vector_type(8))) float v8f;

#define SIGMA3 0.3183098861837907f   // 1/pi
#define TT 4

// ---------------------------------------------------------------------------
// Stage 1: per-edge symmetric-strain scatter.
// strain[(i*4+t)*6 + c] += sym contribution, c = [e00,e11,e22,2e01,2e12,2e20]
// ---------------------------------------------------------------------------
__global__ __launch_bounds__(256) void sph_edge_kernel(
    const float* __restrict__ coords,   // (N,4,3)
    const float* __restrict__ u,        // (N,4,3)
    const float* __restrict__ vol,      // (N,1)
    const int*   __restrict__ eidx,     // (2,E)
    float* __restrict__ strain,         // (N,4,6) accumulator
    int E)
{
    int e = blockIdx.x * blockDim.x + threadIdx.x;
    if (e >= E) return;

    // stream-ahead prefetch of the edge list (global_prefetch_b8)
    __builtin_prefetch(eidx + e + 16384, 0, 0);
    __builtin_prefetch(eidx + E + e + 16384, 0, 0);

    int i = eidx[e];
    int j = eidx[E + e];
    float vj = vol[j];

    const float* ci = coords + (size_t)i * (TT * 3);
    const float* cj = coords + (size_t)j * (TT * 3);
    const float* ui = u + (size_t)i * (TT * 3);
    const float* uj = u + (size_t)j * (TT * 3);
    float* si = strain + (size_t)i * (TT * 6);

#pragma unroll
    for (int t = 0; t < TT; ++t) {
        float rx = cj[t * 3 + 0] - ci[t * 3 + 0];
        float ry = cj[t * 3 + 1] - ci[t * 3 + 1];
        float rz = cj[t * 3 + 2] - ci[t * 3 + 2];
        float rn = sqrtf(rx * rx + ry * ry + rz * rz);
        float rs = fmaxf(rn, 1e-8f);
        float q  = rs;                       // h = 1
        float dW_dq;
        if (q < 1.0f) {
            dW_dq = q * (2.25f * q - 3.0f);
        } else if (q < 2.0f) {
            float w = 2.0f - q;
            dW_dq = -0.75f * w * w;
        } else {
            dW_dq = 0.0f;
        }
        float dW_dr = SIGMA3 * dW_dq;        // h = 1
        float coef  = -dW_dr / rs;           // grad_W = coef * r_vec
        float gx = coef * rx, gy = coef * ry, gz = coef * rz;

        float dux = uj[t * 3 + 0] - ui[t * 3 + 0];
        float duy = uj[t * 3 + 1] - ui[t * 3 + 1];
        float duz = uj[t * 3 + 2] - ui[t * 3 + 2];

        float s0 = vj * dux * gx;                  // eps00
        float s1 = vj * duy * gy;                  // eps11
        float s2 = vj * duz * gz;                  // eps22
        float s3 = vj * (dux * gy + duy * gx);     // 2*eps01
        float s4 = vj * (duy * gz + duz * gy);     // 2*eps12
        float s5 = vj * (duz * gx + dux * gz);     // 2*eps20

        float* d = si + t * 6;
        unsafeAtomicAdd(d + 0, s0);
        unsafeAtomicAdd(d + 1, s1);
        unsafeAtomicAdd(d + 2, s2);
        unsafeAtomicAdd(d + 3, s3);
        unsafeAtomicAdd(d + 4, s4);
        unsafeAtomicAdd(d + 5, s5);
    }
}

// ---------------------------------------------------------------------------
// Stage 2: per-node constitutive quadratic form + external work, block-reduced
// to 8 channels: [0..3] = sum(dc*vol) per t, [4..7] = sum(ext_work*vol) per t.
// ---------------------------------------------------------------------------
__global__ __launch_bounds__(256) void sph_node_kernel(
    const float* __restrict__ strain,   // (N,4,6)
    const float* __restrict__ props,    // (N,4,2)  [E, nu]
    const float* __restrict__ vol,      // (N,1)
    const float* __restrict__ bcv,      // (N,4,3)
    const int*   __restrict__ bct,      // (N,4)
    const float* __restrict__ u,        // (N,4,3)
    float* __restrict__ partial,        // (gridDim.x, 8)
    int N)
{
    __shared__ float red[256 * 8];
    int n   = blockIdx.x * blockDim.x + threadIdx.x;
    int tid = threadIdx.x;

    float ch[8];
#pragma unroll
    for (int c = 0; c < 8; ++c) ch[c] = 0.0f;

    if (n < N) {
        float vn = vol[n];
#pragma unroll
        for (int t = 0; t < TT; ++t) {
            size_t idx = (size_t)n * TT + t;
            const float* s = strain + idx * 6;
            float e1 = s[0], e2 = s[1], e3 = s[2];
            float g1 = s[3], g2 = s[4], g3 = s[5];
            float Em = props[idx * 2 + 0];
            float nu = props[idx * 2 + 1];
            float om2 = 1.0f - 2.0f * nu;
            float lam = Em / ((1.0f + nu) * om2);
            float tr  = e1 + e2 + e3;
            float Q1  = e1 * e1 + e2 * e2 + e3 * e3;
            float Q2  = g1 * g1 + g2 * g2 + g3 * g3;
            float dc  = lam * (om2 * Q1 + nu * tr * tr + 0.5f * om2 * Q2);
            ch[t] = dc * vn;

            float mask = (bct[idx] == 2) ? 1.0f : 0.0f;
            const float* bv = bcv + idx * 3;
            const float* uv = u + idx * 3;
            float ew = mask * (bv[0] * uv[0] + bv[1] * uv[1] + bv[2] * uv[2]);
            ch[4 + t] = ew * vn;
        }
    }

#pragma unroll
    for (int c = 0; c < 8; ++c) red[tid * 8 + c] = ch[c];
    __syncthreads();

    for (int s = 128; s > 0; s >>= 1) {
        if (tid < s) {
#pragma unroll
            for (int c = 0; c < 8; ++c)
                red[tid * 8 + c] += red[(tid + s) * 8 + c];
        }
        __syncthreads();
    }

    if (tid < 8) partial[blockIdx.x * 8 + tid] = red[tid];
}

// ---------------------------------------------------------------------------
// Stage 3: one-wave final reduction using V_WMMA_F32_16X16X4_F32.
// A (16x4): rows m = channel (0..7 valid, 8..15 zero-pad), K = block slice.
// ISA A-layout: lanes 0-15 hold {K0,K1} in {VGPR0,VGPR1}; lanes 16-31 hold
// {K2,K3}. B = all-ones -> C[m,n] = sum_k A[m,k]; lane 0's 8 C VGPRs hold the
// 8 channel totals. EXEC is all-1s (single full wave, converged).
// ---------------------------------------------------------------------------
__global__ __launch_bounds__(32) void sph_final_kernel(
    const float* __restrict__ partial,  // (G,8)
    int G,
    float* __restrict__ out)
{
    int lane = threadIdx.x;
    int m  = lane & 15;                 // channel row
    int k0 = (lane < 16) ? 0 : 2;       // this half-wave's K pair

    float a0 = 0.0f, a1 = 0.0f;
    if (m < 8) {
        for (int g = k0; g < G; g += 4)     a0 += partial[g * 8 + m];
        for (int g = k0 + 1; g < G; g += 4) a1 += partial[g * 8 + m];
    }

    v2f A; A[0] = a0; A[1] = a1;
    v2f B; B[0] = 1.0f; B[1] = 1.0f;
    v8f C = {};
    v8f D = __builtin_amdgcn_wmma_f32_16x16x4_f32(
        false, A, false, B, (short)0, C, false, false);

    if (lane == 0) {
        float li = 0.0f, le = 0.0f;
#pragma unroll
        for (int t = 0; t < TT; ++t) {
            li += 0.5f * fmaxf(D[t], 0.0f);   // pi_int_t
            le += -D[4 + t];                  // pi_ext_t
        }
        out[0] = 0.25f * li + 0.25f * le;     // mean over T=4
    }
}

// ---------------------------------------------------------------------------
extern "C" void kernel_launch(void* const* d_in, const int* in_sizes, int n_in,
                              void* d_out, int out_size, void* d_ws, size_t ws_size,
                              hipStream_t stream) {
    const float* disp   = (const float*)d_in[0];   // (N,4,3)
    const float* coords = (const float*)d_in[1];   // (N,4,3)
    const float* props  = (const float*)d_in[2];   // (N,4,2)
    const float* vol    = (const float*)d_in[3];   // (N,1)
    const float* bcv    = (const float*)d_in[4];   // (N,4,3)
    /* d_in[5] = domain_id (unused) */
    const int*   bct    = (const int*)d_in[6];     // (N,4)
    const int*   eidx   = (const int*)d_in[7];     // (2,E)

    int N = in_sizes[3];            // volume is (N,1)
    int E = in_sizes[7] / 2;

    float* strain  = (float*)d_ws;                     // N*4*6 floats
    int    G       = (N + 255) / 256;
    float* partial = strain + (size_t)N * TT * 6;      // G*8 floats

    size_t zero_bytes = ((size_t)N * TT * 6 + (size_t)G * 8) * sizeof(float);
    hipMemsetAsync(d_ws, 0, zero_bytes, stream);

    sph_edge_kernel<<<(E + 255) / 256, 256, 0, stream>>>(
        coords, disp, vol, eidx, strain, E);

    sph_node_kernel<<<G, 256, 0, stream>>>(
        strain, props, vol, bcv, bct, disp, partial, N);

    sph_final_kernel<<<1, 32, 0, stream>>>(partial, G, (float*)d_out);
}